// SpanModel_8306466750601
// MI455X (gfx1250) — compile-verified
//
#include <hip/hip_runtime.h>
#include <hip/hip_bf16.h>

// ---------------------------------------------------------------------------
// SpanModel preprocessing for MI455X (gfx1250, wave32).
//
// Roofline: ~487MB written + ~104MB read @ 23.3 TB/s HBM => ~25us. Zero matmul
// FLOPs in the reference, so the job is pure data movement. The dominant cost
// (400MB) is the head_rep/tail_rep fan-out: out[b,p,:] = span_rep[b, idx, :],
// idx in [0,128). That is a one-hot(16x128) x span_rep(128x768) product, which
// is EXACT in f32 WMMA (mult by 0/1, accumulate zeros). We run it on the
// matrix pipe (v_wmma_f32_16x16x4_f32), staging 128x16 span_rep column slices
// in LDS (8KB/block, read once) and streaming 16x16 f32 D-tiles to HBM.
// ---------------------------------------------------------------------------

#define Bc  8
#define Sc  4096
#define Dc  768
#define Ec  128
#define Pc  8128      // E*(E-1)/2
#define MPc 25
#define MTc 3500

// Output offsets (flat f32, reference tuple order)
constexpr size_t O0 = 0;                               // prompts_embedding (B,25,768)
constexpr size_t O1 = O0 + (size_t)Bc * MPc * Dc;      // prompts_mask (B,25)
constexpr size_t O2 = O1 + (size_t)Bc * MPc;           // words_embedding (B,3500,768)
constexpr size_t O3 = O2 + (size_t)Bc * MTc * Dc;      // word_mask (B,3500)
constexpr size_t O4 = O3 + (size_t)Bc * MTc;           // pair_idx (B,P,2)
constexpr size_t O5 = O4 + (size_t)Bc * Pc * 2;        // pair_mask (B,P)
constexpr size_t O6 = O5 + (size_t)Bc * Pc;            // head_rep (B,P,768)
constexpr size_t O7 = O6 + (size_t)Bc * Pc * Dc;       // tail_rep (B,P,768)

// Workspace layout (int32 elements)
constexpr int WS_NSPEC = 0;                 // [B]
constexpr int WS_SPOS  = WS_NSPEC + Bc;     // [B][32] special token positions
constexpr int WS_WSRC  = WS_SPOS + Bc * 32; // [B][3500] word target -> source pos (-1 = none)
constexpr int WS_HEAD  = WS_WSRC + Bc * MTc;     // [B][P] head entity idx (0 if unsel)
constexpr int WS_TAIL  = WS_HEAD + Bc * Pc;      // [B][P] tail entity idx (0 if unsel)

typedef __attribute__((ext_vector_type(2))) float v2f;
typedef __attribute__((ext_vector_type(8))) float v8f;

// -------------------------------------------------------------------------
// Kernel 1: per-batch metadata.
//  - special token (id==1) positions -> spos, count -> nspec (scatter-add has
//    unique cumsum slots, so it's a pure gather inversion)
//  - words_mask inverse map: wsrc[b][w-1] = s (unique targets by construction)
//  - prompts_mask and word_mask outputs
// -------------------------------------------------------------------------
__global__ void __launch_bounds__(256) meta_kernel(const int* __restrict__ input_ids,
                                                   const int* __restrict__ words_mask,
                                                   const int* __restrict__ text_len,
                                                   float* __restrict__ out,
                                                   int* __restrict__ nspec,
                                                   int* __restrict__ spos,
                                                   int* __restrict__ wsrc) {
  const int b = blockIdx.x, tid = threadIdx.x;
  // words inverse map
  for (int i = tid; i < MTc; i += 256) wsrc[b * MTc + i] = -1;
  __syncthreads();
  for (int s = tid; s < Sc; s += 256) {
    int w = words_mask[b * Sc + s];
    if (w > 0 && w <= MTc) wsrc[b * MTc + (w - 1)] = s;
  }
  // word mask
  const int L = text_len[b];
  for (int t = tid; t < MTc; t += 256)
    out[O3 + (size_t)b * MTc + t] = (t < L) ? 1.0f : 0.0f;

  // special tokens: block-wide stable rank via prefix sum over per-thread counts
  __shared__ int cnts[256];
  __shared__ int offs[257];
  const int per = Sc / 256; // 16
  const int s0 = tid * per;
  int cnt = 0;
  for (int s = s0; s < s0 + per; ++s) cnt += (input_ids[b * Sc + s] == 1);
  cnts[tid] = cnt;
  __syncthreads();
  if (tid == 0) {
    int acc = 0;
    for (int i = 0; i < 256; ++i) { offs[i] = acc; acc += cnts[i]; }
    offs[256] = acc;
  }
  __syncthreads();
  const int ns = offs[256];
  if (tid == 0) nspec[b] = ns;
  int rank = offs[tid];
  for (int s = s0; s < s0 + per; ++s) {
    if (input_ids[b * Sc + s] == 1) {
      if (rank < 32) spos[b * 32 + rank] = s;
      ++rank;
    }
  }
  for (int k = tid; k < MPc; k += 256)
    out[O1 + (size_t)b * MPc + k] = (k < ns) ? 1.0f : 0.0f;
}

// triu_indices(E,1) row-major inversion: off(r) = r*(2E-1-r)/2
__device__ __forceinline__ int triu_off(int r) { return (r * (2 * Ec - 1 - r)) >> 1; }
__device__ __forceinline__ void rc_from_p(int p, int* rr, int* cc) {
  float t = sqrtf((float)((2 * Ec - 1) * (2 * Ec - 1) - 8 * p));
  int r = (int)((2 * Ec - 1 - t) * 0.5f);
  if (r > Ec - 2) r = Ec - 2;
  if (r < 0) r = 0;
  while (r + 1 <= Ec - 2 && triu_off(r + 1) <= p) ++r;
  while (r > 0 && triu_off(r) > p) --r;
  *rr = r;
  *cc = p - triu_off(r) + r + 1;
}

// -------------------------------------------------------------------------
// Kernel 2: pair selection. Stable argsort of ~sel == stable partition:
// selected pairs keep triu order first, then unselected. slot via prefix sum.
// -------------------------------------------------------------------------
__global__ void __launch_bounds__(256) pairs_kernel(const float* __restrict__ adj,
                                                    float* __restrict__ out,
                                                    int* __restrict__ hd,
                                                    int* __restrict__ tl) {
  const int b = blockIdx.x, tid = threadIdx.x;
  __shared__ int cnts[256];
  __shared__ int offs[257];
  const int per = (Pc + 255) / 256; // 32
  const int p0 = tid * per;
  const int p1 = min(p0 + per, Pc);
  int cnt = 0;
  for (int p = p0; p < p1; ++p) {
    int r, c;
    rc_from_p(p, &r, &c);
    cnt += (adj[(size_t)b * Ec * Ec + (size_t)r * Ec + c] > 0.5f);
  }
  cnts[tid] = cnt;
  __syncthreads();
  if (tid == 0) {
    int acc = 0;
    for (int i = 0; i < 256; ++i) { offs[i] = acc; acc += cnts[i]; }
    offs[256] = acc;
  }
  __syncthreads();
  const int numSel = offs[256];
  int rank = offs[tid]; // # selected among p' < p
  for (int p = p0; p < p1; ++p) {
    int r, c;
    rc_from_p(p, &r, &c);
    const bool sel = adj[(size_t)b * Ec * Ec + (size_t)r * Ec + c] > 0.5f;
    const int slot = sel ? rank : (numSel + (p - rank));
    if (sel) ++rank;
    out[O4 + ((size_t)b * Pc + slot) * 2 + 0] = sel ? (float)r : -1.0f;
    out[O4 + ((size_t)b * Pc + slot) * 2 + 1] = sel ? (float)c : -1.0f;
    out[O5 + (size_t)b * Pc + slot] = sel ? 1.0f : 0.0f;
    hd[b * Pc + slot] = sel ? r : 0;
    tl[b * Pc + slot] = sel ? c : 0;
  }
}

// -------------------------------------------------------------------------
// Kernel 3/4: row gathers for prompts_embedding and words_embedding.
// 192 threads = one 768-f32 row as float4 (3KB streaming stores).
// -------------------------------------------------------------------------
__global__ void __launch_bounds__(192) prompt_rows(const float* __restrict__ te,
                                                   const int* __restrict__ nspec,
                                                   const int* __restrict__ spos,
                                                   float* __restrict__ out) {
  const int row = blockIdx.x, b = row / MPc, k = row % MPc, tid = threadIdx.x;
  float4 v = make_float4(0.f, 0.f, 0.f, 0.f);
  if (k < nspec[b]) {
    const int s = spos[b * 32 + k];
    v = *(const float4*)(te + ((size_t)b * Sc + s) * Dc + tid * 4);
  }
  *(float4*)(out + O0 + ((size_t)b * MPc + k) * Dc + tid * 4) = v;
}

__global__ void __launch_bounds__(192) word_rows(const float* __restrict__ te,
                                                 const int* __restrict__ wsrc,
                                                 float* __restrict__ out) {
  const int row = blockIdx.x, b = row / MTc, t = row % MTc, tid = threadIdx.x;
  const int s = wsrc[b * MTc + t];
  float4 v = make_float4(0.f, 0.f, 0.f, 0.f);
  if (s >= 0) v = *(const float4*)(te + ((size_t)b * Sc + s) * Dc + tid * 4);
  *(float4*)(out + O2 + ((size_t)b * MTc + t) * Dc + tid * 4) = v;
}

// -------------------------------------------------------------------------
// Kernel 5: head/tail_rep via one-hot f32 WMMA (exact: x*1/x*0 + 0-accum).
// Block = (batch b, 16-col slice nt, m-chunk). Stages span_rep[b][:,16 cols]
// in LDS once (8KB), then each wave emits 16x16 D tiles:
//   D = OneHot(idx[16] over K=128) x Bt(128x16),  K-step 4, 32 wmma/tile.
// Layouts per CDNA5 ISA (16x16x4 F32):
//   A/B: lane half h=lane>>4, M/N=lane&15, component v -> K = 4*step + v + 2h
//   C/D: vgpr j, half h -> M = j + 8h, N = lane&15
// -------------------------------------------------------------------------
__global__ void __launch_bounds__(256) pair_rep_wmma(const float* __restrict__ span,
                                                     const int* __restrict__ hd,
                                                     const int* __restrict__ tl,
                                                     float* __restrict__ out) {
  const int b = blockIdx.x / (Dc / 16);
  const int nt = blockIdx.x % (Dc / 16);
  const int chunk = blockIdx.y; // 4 chunks of m-side tiles
  const int tid = threadIdx.x;

  __shared__ float Bt[Ec][16];
  for (int i = tid; i < Ec * 16; i += 256) {
    const int row = i >> 4, col = i & 15;
    Bt[row][col] = span[((size_t)b * Ec + row) * Dc + nt * 16 + col];
  }
  __syncthreads();

  const int lane = tid & 31;
  const int wave = tid >> 5;
  const int h = lane >> 4;     // lane half
  const int col = lane & 15;   // M for A / N for B,D

  const int TILES = (Pc / 16) * 2;      // 508 m-tiles x {head,tail} = 1016
  const int t0 = chunk * (TILES / 4) + wave;
  const int t1 = (chunk + 1) * (TILES / 4);

  for (int t = t0; t < t1; t += 8) {    // uniform trip count within a wave
    const int side = t & 1;
    const int m = t >> 1;
    const int* __restrict__ idxArr = side ? tl : hd;
    const int idx = idxArr[b * Pc + m * 16 + col]; // entity row for output row M=col

    v8f c = {0.f, 0.f, 0.f, 0.f, 0.f, 0.f, 0.f, 0.f};
#pragma unroll
    for (int s = 0; s < Ec / 4; ++s) {
      const int k = 4 * s + 2 * h;
      v2f a, bb;
      a.x = (idx == k) ? 1.0f : 0.0f;
      a.y = (idx == k + 1) ? 1.0f : 0.0f;
      bb.x = Bt[k][col];
      bb.y = Bt[k + 1][col];
      c = __builtin_amdgcn_wmma_f32_16x16x4_f32(false, a, false, bb, (short)0, c,
                                                false, false);
    }

    const size_t base = (side ? O7 : O6) + ((size_t)b * Pc + (size_t)m * 16) * Dc +
                        (size_t)nt * 16 + col;
#pragma unroll
    for (int j = 0; j < 8; ++j)
      out[base + (size_t)(j + 8 * h) * Dc] = c[j];
  }
}

extern "C" void kernel_launch(void* const* d_in, const int* in_sizes, int n_in,
                              void* d_out, int out_size, void* d_ws, size_t ws_size,
                              hipStream_t stream) {
  const float* te       = (const float*)d_in[0]; // token_embeds (B,S,D)
  const float* adj      = (const float*)d_in[1]; // adj (B,E,E)
  const float* span     = (const float*)d_in[2]; // span_rep (B,E,D)
  const int* input_ids  = (const int*)d_in[3];
  // d_in[4] attention_mask: unused by the reference outputs
  const int* text_len   = (const int*)d_in[5];
  const int* words_mask = (const int*)d_in[6];

  float* out = (float*)d_out;
  int* ws = (int*)d_ws;
  int* nspec = ws + WS_NSPEC;
  int* spos  = ws + WS_SPOS;
  int* wsrc  = ws + WS_WSRC;
  int* hd    = ws + WS_HEAD;
  int* tl    = ws + WS_TAIL;

  meta_kernel<<<Bc, 256, 0, stream>>>(input_ids, words_mask, text_len, out, nspec,
                                      spos, wsrc);
  pairs_kernel<<<Bc, 256, 0, stream>>>(adj, out, hd, tl);
  prompt_rows<<<Bc * MPc, 192, 0, stream>>>(te, nspec, spos, out);
  word_rows<<<Bc * MTc, 192, 0, stream>>>(te, wsrc, out);
  pair_rep_wmma<<<dim3(Bc * (Dc / 16), 4), 256, 0, stream>>>(span, hd, tl, out);
}